// Conv2d_bilinear_v_41403484733516
// MI455X (gfx1250) — compile-verified
//
#include <hip/hip_runtime.h>
#include <hip/hip_bf16.h>

typedef __attribute__((ext_vector_type(2))) float v2f;
typedef __attribute__((ext_vector_type(8))) float v8f;

#define NB   4
#define CCH  96
#define HH   32
#define WW   32
#define LL   (HH * WW)          // 1024
#define OO   96
#define CC2  (CCH * CCH)        // 9216

// ---------------------------------------------------------------------------
// Kernel 1: 3x3 zero-padded average pool, output layout p[n][l][c] (c fastest)
// ---------------------------------------------------------------------------
__global__ void pool_kernel(const float* __restrict__ x, float* __restrict__ p) {
    int idx = blockIdx.x * blockDim.x + threadIdx.x;
    const int total = NB * LL * CCH;
    if (idx >= total) return;
    int c = idx % CCH;
    int l = (idx / CCH) & (LL - 1);
    int n = idx / (CCH * LL);
    int h = l >> 5;
    int w = l & 31;
    const float* xc = x + ((n * CCH + c) * HH) * WW;
    float s = 0.0f;
    #pragma unroll
    for (int dh = -1; dh <= 1; ++dh) {
        int hh = h + dh;
        if (hh < 0 || hh >= HH) continue;
        #pragma unroll
        for (int dw = -1; dw <= 1; ++dw) {
            int ww2 = w + dw;
            if (ww2 < 0 || ww2 >= WW) continue;
            s += xc[hh * WW + ww2];
        }
    }
    p[idx] = s * (1.0f / 9.0f);
}

// ---------------------------------------------------------------------------
// Kernel 2: masked weight, transposed: WMT[o][d][c] = weight[o][c][d]*mask[o][c][d]
// Transpose makes (k,k+1) contiguous -> B fragment is one float2 load per lane.
// ---------------------------------------------------------------------------
__global__ void wmt_kernel(const float* __restrict__ weight,
                           const float* __restrict__ mask,
                           float* __restrict__ wmt) {
    int idx = blockIdx.x * blockDim.x + threadIdx.x;
    const int total = OO * CC2;
    if (idx >= total) return;
    int o = idx / CC2;
    int r = idx % CC2;
    int c = r / CCH;
    int d = r % CCH;
    wmt[o * CC2 + d * CCH + c] = weight[idx] * mask[idx];
}

// ---------------------------------------------------------------------------
// Kernel 3: per (o, n): T = P(16xC) @ WM_o(CxC) via V_WMMA_F32_16X16X4_F32,
// then y[l] = sum_d T[l,d] * P[l,d], + bias.
// grid = (O, N); block = 256 (8 waves); each wave handles 8 L-tiles of 16 rows.
// ---------------------------------------------------------------------------
__global__ void __launch_bounds__(256)
bilinear_wmma_kernel(const float* __restrict__ P,
                     const float* __restrict__ WMT,
                     const float* __restrict__ bias,
                     float* __restrict__ out) {
    const int o = blockIdx.x;
    const int n = blockIdx.y;

    const float* Pn = P + n * (LL * CCH);
    const float* Wo = WMT + o * CC2;

    const int lane = threadIdx.x & 31;
    const int wave = threadIdx.x >> 5;
    const int m    = lane & 15;      // A: row m   | B: col n  | C/D: col n
    const int half = lane >> 4;      // k sub-pair / row half select

    // B-fragment base: lane reads WMT[o][d0+m][ks], WMT[o][d0+m][ks+1]
    // (contiguous thanks to the transpose); ks = 4*k + 2*half.
    const float bo = bias[o];

    for (int t = wave; t < LL / 16; t += 8) {
        const int l0 = t * 16;

        // ---- load all 24 A fragments for this L-tile (row m, k = 0..95) ----
        v2f a[24];
        const float* prow = Pn + (l0 + m) * CCH + 2 * half;
        #pragma unroll
        for (int k = 0; k < 24; ++k)
            a[k] = *(const v2f*)(prow + k * 4);

        float yp[8];
        #pragma unroll
        for (int j = 0; j < 8; ++j) yp[j] = 0.0f;

        #pragma unroll
        for (int dt = 0; dt < 6; ++dt) {
            v8f acc = {};
            const float* wrow = Wo + (dt * 16 + m) * CCH + 2 * half;
            #pragma unroll
            for (int k = 0; k < 24; ++k) {
                v2f b = *(const v2f*)(wrow + k * 4);
                // D = A(16x4 f32) * B(4x16 f32) + C(16x16 f32)
                acc = __builtin_amdgcn_wmma_f32_16x16x4_f32(
                        /*neg_a=*/false, a[k],
                        /*neg_b=*/false, b,
                        /*c_mod=*/(short)0, acc,
                        /*reuse_a=*/false, /*reuse_b=*/false);
            }
            // C/D layout: VGPR j holds row (j + 8*half), column = lane%16.
            // Row-dot partial: yp[j] += T[row][d0+m_col] * P[row][d0+m_col]
            const float* pd = Pn + (l0 + 8 * half) * CCH + dt * 16 + m;
            #pragma unroll
            for (int j = 0; j < 8; ++j)
                yp[j] += acc[j] * pd[j * CCH];
        }

        // reduce over the 16 column-lanes within each half-wave
        #pragma unroll
        for (int j = 0; j < 8; ++j) {
            float v = yp[j];
            v += __shfl_xor(v, 1, 16);
            v += __shfl_xor(v, 2, 16);
            v += __shfl_xor(v, 4, 16);
            v += __shfl_xor(v, 8, 16);
            yp[j] = v;
        }

        if (m == 0) {
            float* orow = out + ((n * OO + o) << 10) + l0 + 8 * half;
            #pragma unroll
            for (int j = 0; j < 8; ++j)
                orow[j] = yp[j] + bo;
        }
    }
}

// ---------------------------------------------------------------------------
extern "C" void kernel_launch(void* const* d_in, const int* in_sizes, int n_in,
                              void* d_out, int out_size, void* d_ws, size_t ws_size,
                              hipStream_t stream) {
    const float* x      = (const float*)d_in[0];  // [N,C,H,W]
    const float* weight = (const float*)d_in[1];  // [O,C,C]
    const float* bias   = (const float*)d_in[2];  // [O]
    const float* mask   = (const float*)d_in[3];  // [O,C,C]
    float* out = (float*)d_out;                   // [N,O,H,W]

    float* P   = (float*)d_ws;                    // [N,L,C]   = 393216 f32
    float* WMT = P + NB * LL * CCH;               // [O,C,C]^T = 884736 f32

    {
        const int total = NB * LL * CCH;
        pool_kernel<<<(total + 255) / 256, 256, 0, stream>>>(x, P);
    }
    {
        const int total = OO * CC2;
        wmt_kernel<<<(total + 255) / 256, 256, 0, stream>>>(weight, mask, WMT);
    }
    {
        dim3 grid(OO, NB);
        bilinear_wmma_kernel<<<grid, 256, 0, stream>>>(P, WMT, bias, out);
    }
}